// NoisyTopkRouter_15659450761991
// MI455X (gfx1250) — compile-verified
//
#include <hip/hip_runtime.h>
#include <math.h>

// Problem constants (from reference setup_inputs)
#define B_SZ 4096
#define C_SZ 512
#define E_SZ 64
#define HW   196
#define HW4  49          // 196 floats = 49 float4 (784 B = 49*16 -> rows 16B aligned)
#define TOPK 8
#define LROW 65          // LDS row stride (floats) -> conflict-free column scans

typedef float v2f __attribute__((ext_vector_type(2)));
typedef float v8f __attribute__((ext_vector_type(8)));

// ---------------------------------------------------------------------------
// Kernel 1: global average pool  [B,C,H,W] -> pooled [B,C]
// One wave32 per (b,c) row of 196 floats; float4 coalesced loads + shfl reduce.
// Bandwidth bound: 1.64 GB read -> ~70us at 23.3 TB/s.
// ---------------------------------------------------------------------------
__global__ __launch_bounds__(256) void pool_kernel(const float* __restrict__ mh,
                                                   float* __restrict__ pooled) {
    const int wave = threadIdx.x >> 5;
    const int lane = threadIdx.x & 31;
    const size_t row = (size_t)blockIdx.x * 8 + wave;   // (b*C + c)

    const float4* __restrict__ src = (const float4*)mh + row * HW4;
    float4 v0 = src[lane];
    float s = v0.x + v0.y + v0.z + v0.w;
    if (lane < HW4 - 32) {                               // lanes 0..16 take 2nd batch
        float4 v1 = src[32 + lane];
        s += v1.x + v1.y + v1.z + v1.w;
    }
    // wave32 butterfly reduction
    #pragma unroll
    for (int off = 16; off >= 1; off >>= 1)
        s += __shfl_xor(s, off, 32);

    if (lane == 0) pooled[row] = s * (1.0f / (float)HW);
}

// ---------------------------------------------------------------------------
// Kernel 2: router. One wave handles 16 batch rows.
//   logits  = pooled @ W_route^T + b_route   (fp32 WMMA 16x16x4, K=512)
//   nlogits = pooled @ W_noise^T + b_noise
// then per-row softmax / softplus-noise-softmax / top-8 / renorm.
// 4 waves / block -> static LDS = 4*2*16*65*4 = 33,280 B.
// ---------------------------------------------------------------------------
__global__ __launch_bounds__(128) void router_kernel(
    const float* __restrict__ pooled, const float* __restrict__ noise,
    const float* __restrict__ Wr, const float* __restrict__ br,
    const float* __restrict__ Wn, const float* __restrict__ bn,
    float* __restrict__ topk_out, int* __restrict__ idx_out,
    float* __restrict__ noisy_out)
{
    __shared__ float route_s[4][16 * LROW];
    __shared__ float noise_s[4][16 * LROW];

    const int wave = threadIdx.x >> 5;
    const int lane = threadIdx.x & 31;
    const int rowBase = (blockIdx.x * 4 + wave) * 16;     // first batch row

    // fp32 WMMA A-frag layout: lanes 0-15 hold (M=lane, K={k,k+1}),
    //                          lanes 16-31 hold (M=lane-16, K={k+2,k+3})
    const int m  = lane & 15;
    const int kh = (lane >> 4) << 1;                      // 0 or 2

    v8f accR[4] = {{}, {}, {}, {}};
    v8f accN[4] = {{}, {}, {}, {}};

    const float* __restrict__ Arow = pooled + (size_t)(rowBase + m) * C_SZ + kh;

    for (int k = 0; k < C_SZ; k += 4) {
        v2f a;
        a.x = Arow[k];
        a.y = Arow[k + 1];
        #pragma unroll
        for (int t = 0; t < 4; ++t) {
            // B-frag (4x16, K x N): B[k][n] = W[e=n][c=k]  (out = pooled @ W^T)
            const float* pr = Wr + (size_t)(t * 16 + m) * C_SZ + kh + k;
            const float* pn = Wn + (size_t)(t * 16 + m) * C_SZ + kh + k;
            v2f bR; bR.x = pr[0]; bR.y = pr[1];
            v2f bN; bN.x = pn[0]; bN.y = pn[1];
            accR[t] = __builtin_amdgcn_wmma_f32_16x16x4_f32(
                false, a, false, bR, (short)0, accR[t], false, false);
            accN[t] = __builtin_amdgcn_wmma_f32_16x16x4_f32(
                false, a, false, bN, (short)0, accN[t], false, false);
        }
    }

    // D layout: VGPR j -> row j (lanes 0-15) / row j+8 (lanes 16-31), col = lane&15
    #pragma unroll
    for (int t = 0; t < 4; ++t) {
        const int e   = t * 16 + m;
        const float bre = br[e];
        const float bne = bn[e];
        #pragma unroll
        for (int j = 0; j < 8; ++j) {
            const int r = (lane < 16) ? j : (j + 8);
            route_s[wave][r * LROW + e] = accR[t][j] + bre;
            noise_s[wave][r * LROW + e] = accN[t][j] + bne;
        }
    }
    __syncthreads();

    // One lane per batch row for the 64-wide gating math.
    if (lane < 16) {
        const int r = lane;
        const int b = rowBase + r;
        float* RS = &route_s[wave][r * LROW];
        float* NS = &noise_s[wave][r * LROW];

        // softmax over route logits
        float mx = -INFINITY;
        for (int e = 0; e < E_SZ; ++e) mx = fmaxf(mx, RS[e]);
        float sum = 0.f;
        for (int e = 0; e < E_SZ; ++e) { float v = expf(RS[e] - mx); RS[e] = v; sum += v; }
        const float inv = 1.0f / sum;

        // noise branch: z = noise * softplus(noise_logits), then softmax(z)
        float mn = -INFINITY;
        for (int e = 0; e < E_SZ; ++e) {
            float x  = NS[e];
            float sp = fmaxf(x, 0.0f) + log1pf(expf(-fabsf(x)));   // stable softplus
            float z  = noise[(size_t)b * E_SZ + e] * sp;
            NS[e] = z;
            mn = fmaxf(mn, z);
        }
        float sumn = 0.f;
        for (int e = 0; e < E_SZ; ++e) { float v = expf(NS[e] - mn); NS[e] = v; sumn += v; }
        const float invn = 1.0f / sumn;

        // noisy logits = softmax(route) + softmax(noise-scaled)
        for (int e = 0; e < E_SZ; ++e) {
            float nz = RS[e] * inv + NS[e] * invn;
            noisy_out[(size_t)b * E_SZ + e] = nz;
            RS[e] = nz;                                   // reuse for top-k
        }

        // top-8 (descending; lowest index wins ties, matching jax.lax.top_k)
        float vals[TOPK]; int ids[TOPK];
        for (int kk = 0; kk < TOPK; ++kk) {
            float best = -INFINITY; int bi = 0;
            for (int e = 0; e < E_SZ; ++e) {
                float v = RS[e];
                if (v > best) { best = v; bi = e; }
            }
            vals[kk] = best; ids[kk] = bi;
            RS[bi] = -INFINITY;
        }

        // softmax over the selected 8 (vals[0] is the max)
        const float mt = vals[0];
        float st = 0.f, ex[TOPK];
        #pragma unroll
        for (int kk = 0; kk < TOPK; ++kk) { ex[kk] = expf(vals[kk] - mt); st += ex[kk]; }
        const float ist = 1.0f / st;
        #pragma unroll
        for (int kk = 0; kk < TOPK; ++kk) {
            topk_out[(size_t)b * TOPK + kk] = ex[kk] * ist;
            idx_out [(size_t)b * TOPK + kk] = ids[kk];
        }
    }
}

// ---------------------------------------------------------------------------
extern "C" void kernel_launch(void* const* d_in, const int* in_sizes, int n_in,
                              void* d_out, int out_size, void* d_ws, size_t ws_size,
                              hipStream_t stream) {
    const float* mh    = (const float*)d_in[0];   // [B,C,H,W]
    const float* noise = (const float*)d_in[1];   // [B,E]
    const float* Wr    = (const float*)d_in[2];   // [E,C]
    const float* br    = (const float*)d_in[3];   // [E]
    const float* Wn    = (const float*)d_in[4];   // [E,C]
    const float* bn    = (const float*)d_in[5];   // [E]

    float* out       = (float*)d_out;
    float* topk_out  = out;                                    // [B,8] f32
    int*   idx_out   = (int*)(out + (size_t)B_SZ * TOPK);      // [B,8] i32
    float* noisy_out = out + (size_t)B_SZ * TOPK * 2;          // [B,64] f32

    float* pooled = (float*)d_ws;                              // [B,C] = 8 MB scratch

    // Pool: one wave per (b,c) row; 8 waves per 256-thread block.
    dim3 g1((B_SZ * C_SZ) / 8), b1(256);
    pool_kernel<<<g1, b1, 0, stream>>>(mh, pooled);

    // Router: 4 waves/block, 16 rows/wave -> 64 rows/block.
    dim3 g2(B_SZ / 64), b2(128);
    router_kernel<<<g2, b2, 0, stream>>>(pooled, noise, Wr, br, Wn, bn,
                                         topk_out, idx_out, noisy_out);
}